// PhysicsRegularizationPACE_18030272709228
// MI455X (gfx1250) — compile-verified
//
#include <hip/hip_runtime.h>
#include <math.h>

#define IMG   256
#define ANG   180
#define DET   256
#define NB    2
#define NC    256
#define SINO_W 1.0f
#define TV_W   0.01f

typedef __attribute__((ext_vector_type(2))) float v2f;
typedef __attribute__((ext_vector_type(8))) float v8f;

// Workspace layout (floats): [0]=sino_acc [1]=dx_acc [2]=dy_acc, [16..] x image (NB*IMG*IMG)

__global__ void zero_acc_kernel(float* ws) {
  if (threadIdx.x < 3) ws[threadIdx.x] = 0.0f;
}

// x[b,h,w] = sum_c w[c] * f[b,c,h,w] via V_WMMA_F32_16X16X4_F32.
// One wave handles 16 consecutive flattened pixels. A-matrix rows are all
// identical (the 4-wide w chunk), so every row of D carries the answer.
// ISA f32 16x4 A layout: lanes 0-15 hold K=0 (vgpr0) / K=1 (vgpr1),
// lanes 16-31 hold K=2 / K=3.  B (4x16) mirrors the K split; N = lane%16.
__global__ __launch_bounds__(256) void conv1x1_wmma_kernel(
    const float* __restrict__ feat, const float* __restrict__ wproj,
    float* __restrict__ xout) {
  const int HW = IMG * IMG;
  int lane = threadIdx.x & 31;
  int wave = threadIdx.x >> 5;
  int tile = blockIdx.x * 8 + wave;      // 16-pixel tile over B*H*W
  int n    = lane & 15;
  int pix  = tile * 16 + n;
  int b    = pix / HW;
  int p    = pix - b * HW;
  const float* fb = feat + (size_t)b * NC * HW + p;
  int hi = lane >> 4;                    // 0 -> K={0,1}, 1 -> K={2,3}

  v8f acc = {};
  #pragma unroll 4
  for (int k0 = 0; k0 < NC; k0 += 4) {
    // wave-uniform -> scalar loads from K$
    float w0 = wproj[k0 + 0];
    float w1 = wproj[k0 + 1];
    float w2 = wproj[k0 + 2];
    float w3 = wproj[k0 + 3];
    v2f a, bm;
    a.x = hi ? w2 : w0;
    a.y = hi ? w3 : w1;
    int ka = k0 + hi * 2;
    bm.x = fb[(size_t)ka * HW];
    bm.y = fb[(size_t)(ka + 1) * HW];
    acc = __builtin_amdgcn_wmma_f32_16x16x4_f32(false, a, false, bm, (short)0,
                                                acc, false, false);
  }
  // All D rows equal; D vgpr0 lane L holds pixel tile*16 + L%16.
  if (lane < 16) xout[tile * 16 + lane] = acc[0];
}

// One block per (b, angle); thread = detector index. Integrate over u,
// bilinear sampling with the reference's align_corners + clip + inside mask.
__global__ __launch_bounds__(256) void radon_loss_kernel(
    const float* __restrict__ x, const float* __restrict__ tgt,
    const float* __restrict__ wts, float* __restrict__ acc) {
  __shared__ float red[8];
  int a = blockIdx.x % ANG;
  int b = blockIdx.x / ANG;
  int t = threadIdx.x;
  float theta = (float)a * (3.14159265358979323846f / (float)ANG);
  float st, ct;
  sincosf(theta, &st, &ct);
  float tc = -1.0f + 2.0f * (float)t / (float)(DET - 1);
  const float* xb = x + b * IMG * IMG;

  float s = 0.0f;
  for (int ui = 0; ui < IMG; ++ui) {
    float u = -1.0f + 2.0f * (float)ui / (float)(IMG - 1);
    float X = tc * ct - u * st;
    float Y = tc * st + u * ct;
    float inside = (X >= -1.0f && X <= 1.0f && Y >= -1.0f && Y <= 1.0f) ? 1.0f : 0.0f;
    float px = (X + 1.0f) * 0.5f * (float)(IMG - 1);
    float py = (Y + 1.0f) * 0.5f * (float)(IMG - 1);
    float fx = floorf(px), fy = floorf(py);
    float wx = px - fx, wy = py - fy;
    int x0 = (int)fx, y0 = (int)fy;
    int x0c = min(max(x0, 0), IMG - 1);
    int x1c = min(max(x0 + 1, 0), IMG - 1);
    int y0c = min(max(y0, 0), IMG - 1);
    int y1c = min(max(y0 + 1, 0), IMG - 1);
    float v00 = xb[y0c * IMG + x0c];
    float v01 = xb[y0c * IMG + x1c];
    float v10 = xb[y1c * IMG + x0c];
    float v11 = xb[y1c * IMG + x1c];
    float val = v00 * (1.0f - wx) * (1.0f - wy) + v01 * wx * (1.0f - wy)
              + v10 * (1.0f - wx) * wy         + v11 * wx * wy;
    s += val * inside;
  }
  int idx = (b * ANG + a) * DET + t;
  float d = s - tgt[idx];
  float c = wts[idx] * d * d;
  for (int off = 16; off > 0; off >>= 1) c += __shfl_down(c, off, 32);
  if ((threadIdx.x & 31) == 0) red[threadIdx.x >> 5] = c;
  __syncthreads();
  if (threadIdx.x == 0) {
    float tot = 0.0f;
    for (int i = 0; i < 8; ++i) tot += red[i];
    atomicAdd(&acc[0], tot);
  }
}

__global__ __launch_bounds__(256) void tv_loss_kernel(
    const float* __restrict__ x, float* __restrict__ acc) {
  __shared__ float redx[8];
  __shared__ float redy[8];
  int idx = blockIdx.x * blockDim.x + threadIdx.x;   // over NB*IMG*IMG
  int b = idx / (IMG * IMG);
  int r = idx - b * (IMG * IMG);
  int h = r / IMG, w = r - h * IMG;
  float v = x[idx];
  float sdx = (w + 1 < IMG) ? fabsf(x[idx + 1]   - v) : 0.0f;
  float sdy = (h + 1 < IMG) ? fabsf(x[idx + IMG] - v) : 0.0f;
  for (int off = 16; off > 0; off >>= 1) {
    sdx += __shfl_down(sdx, off, 32);
    sdy += __shfl_down(sdy, off, 32);
  }
  if ((threadIdx.x & 31) == 0) {
    redx[threadIdx.x >> 5] = sdx;
    redy[threadIdx.x >> 5] = sdy;
  }
  __syncthreads();
  if (threadIdx.x == 0) {
    float tx = 0.0f, ty = 0.0f;
    for (int i = 0; i < 8; ++i) { tx += redx[i]; ty += redy[i]; }
    atomicAdd(&acc[1], tx);
    atomicAdd(&acc[2], ty);
  }
}

__global__ void finalize_kernel(const float* __restrict__ acc, float* __restrict__ out) {
  if (threadIdx.x == 0 && blockIdx.x == 0) {
    float sino = acc[0] / (float)(NB * ANG * DET);
    float dx   = acc[1] / (float)(NB * IMG * (IMG - 1));
    float dy   = acc[2] / (float)(NB * (IMG - 1) * IMG);
    out[0] = SINO_W * sino + TV_W * (dx + dy);
  }
}

extern "C" void kernel_launch(void* const* d_in, const int* in_sizes, int n_in,
                              void* d_out, int out_size, void* d_ws, size_t ws_size,
                              hipStream_t stream) {
  (void)in_sizes; (void)n_in; (void)out_size; (void)ws_size;
  const float* feat = (const float*)d_in[0];   // [B,C,IMG,IMG]
  const float* wprj = (const float*)d_in[1];   // [C]
  const float* tgt  = (const float*)d_in[2];   // [B,1,ANG,DET]
  const float* wts  = (const float*)d_in[3];   // [B,1,ANG,DET]
  float* out = (float*)d_out;
  float* ws  = (float*)d_ws;
  float* ximg = ws + 16;                       // NB*IMG*IMG floats

  zero_acc_kernel<<<1, 32, 0, stream>>>(ws);

  // 8192 16-pixel tiles, 8 waves (tiles) per 256-thread block
  conv1x1_wmma_kernel<<<(NB * IMG * IMG) / 128, 256, 0, stream>>>(feat, wprj, ximg);

  radon_loss_kernel<<<NB * ANG, DET, 0, stream>>>(ximg, tgt, wts, ws);

  tv_loss_kernel<<<(NB * IMG * IMG) / 256, 256, 0, stream>>>(ximg, ws);

  finalize_kernel<<<1, 32, 0, stream>>>(ws, out);
}